// CoocOpModel_67800353735082
// MI455X (gfx1250) — compile-verified
//
#include <hip/hip_runtime.h>

// Problem constants from the reference: B=64, V=8192, K=3 op channels.
#define B_DIM 64
#define V_DIM 8192
#define STR   (V_DIM + 2)      // func/arg row stride (V + 2 flag columns)
#define QINF  1000000000.0f

typedef __attribute__((ext_vector_type(2))) float v2f;   // WMMA f32 A/B operand (16x4 / 4x16)
typedef __attribute__((ext_vector_type(8))) float v8f;   // WMMA f32 C/D accumulator (16x16)

struct F3 { float x, y, z; };  // one (i,j) cell of C: all 3 k values, 12 contiguous bytes

// out[b,k] = mask[b,k] (computed from the two flag columns of func); main kernel atomically adds the einsum.
__global__ void cooc_mask_init(const float* __restrict__ func, float* __restrict__ out) {
  int b = threadIdx.x;
  if (b < B_DIM) {
    float a1 = func[(size_t)b * STR + V_DIM + 0];
    float a2 = func[(size_t)b * STR + V_DIM + 1];
    out[b * 3 + 0] = a1 * -QINF;
    out[b * 3 + 1] = (1.0f - (1.0f - a1) * (1.0f - a2)) * -QINF;
    out[b * 3 + 2] = 0.0f;
  }
}

// T_k[b,j] = sum_i func[b,i] * C[i,j,k]   (WMMA: M=b(16), K=i(4/step), N=j(16), 3 accumulators for k)
// out[b,k] += sum_j T_k[b,j] * arg[b,j]   (cross-lane reduce + atomic)
// Block = 4 waves (one per 16-row b tile, sharing the C stream via WGP$); grid = V/16 j-tiles.
__global__ __launch_bounds__(128) void cooc_einsum_wmma(
    const float* __restrict__ func,
    const float* __restrict__ arg,
    const float* __restrict__ cooc,
    float* __restrict__ out) {
  const int lane = threadIdx.x & 31;
  const int hh   = lane >> 4;          // half-wave select (A rows K=0,1 vs K=2,3; acc M or M+8)
  const int l15  = lane & 15;
  const int b0   = (threadIdx.x >> 5) << 4;   // 16-row b tile per wave
  const int j0   = (int)blockIdx.x << 4;      // 16-col j tile per block

  // A operand: func[b0+l15, i + 2*hh .. +1]  (16x4 f32 layout: lanes 0-15 K=0,1; lanes 16-31 K=2,3)
  const float* aPtr = func + (size_t)(b0 + l15) * STR + (hh << 1);
  // B operand: C[(i + 2*hh)*V + (j0+l15)]*3  (4x16 f32 layout: v0=row K(2*hh), v1=row K+1, cols = l15)
  const float* bPtr = cooc + ((size_t)(hh << 1) * V_DIM + (size_t)(j0 + l15)) * 3;

  v8f acc0 = {}, acc1 = {}, acc2 = {};   // one 16x16 f32 accumulator per k channel

  #pragma unroll 8
  for (int i = 0; i < V_DIM; i += 4) {
    if ((i & 31) == 0) {
      // C stream is strided (96KB row stride): prefetch ~8 steps (32 i-rows) ahead for both rows.
      __builtin_prefetch(bPtr + (size_t)12 * V_DIM * 8, 0, 0);
      __builtin_prefetch(bPtr + (size_t)12 * V_DIM * 8 + 3 * V_DIM, 0, 0);
    }
    v2f a  = *(const v2f*)aPtr;                    // func[b, i+2hh], func[b, i+2hh+1] (8B aligned)
    F3  r0 = *(const F3*)bPtr;                     // C[i+2hh,   j, 0..2]
    F3  r1 = *(const F3*)(bPtr + 3 * V_DIM);       // C[i+2hh+1, j, 0..2]
    aPtr += 4;
    bPtr += (size_t)12 * V_DIM;                    // advance 4 i-rows
    v2f bv0 = { r0.x, r1.x };                      // k=0 column slice
    v2f bv1 = { r0.y, r1.y };                      // k=1
    v2f bv2 = { r0.z, r1.z };                      // k=2
    // Three WMMAs share A: 3x2048 FLOP per 768B of fresh C data per wave (x4 b-tiles sharing => 32 flop/B).
    acc0 = __builtin_amdgcn_wmma_f32_16x16x4_f32(false, a, false, bv0, (short)0, acc0, false, false);
    acc1 = __builtin_amdgcn_wmma_f32_16x16x4_f32(false, a, false, bv1, (short)0, acc1, false, false);
    acc2 = __builtin_amdgcn_wmma_f32_16x16x4_f32(false, a, false, bv2, (short)0, acc2, false, false);
  }

  // Contract the 16x16 tile with arg over j (the 16 lanes of each half), then atomically accumulate.
  #pragma unroll
  for (int r = 0; r < 8; ++r) {
    int b = b0 + (hh << 3) + r;                            // acc VGPR r -> M = r (+8 for upper lanes)
    float av = arg[(size_t)b * STR + j0 + l15];            // arg[b, j0+l15]
    float s0 = acc0[r] * av;
    float s1 = acc1[r] * av;
    float s2 = acc2[r] * av;
    #pragma unroll
    for (int m = 1; m < 16; m <<= 1) {                     // reduce within each 16-lane half (j dim)
      s0 += __shfl_xor(s0, m, 32);
      s1 += __shfl_xor(s1, m, 32);
      s2 += __shfl_xor(s2, m, 32);
    }
    if (l15 == 0) {
      atomicAdd(&out[b * 3 + 0], s0);
      atomicAdd(&out[b * 3 + 1], s1);
      atomicAdd(&out[b * 3 + 2], s2);
    }
  }
}

extern "C" void kernel_launch(void* const* d_in, const int* in_sizes, int n_in,
                              void* d_out, int out_size, void* d_ws, size_t ws_size,
                              hipStream_t stream) {
  const float* func = (const float*)d_in[0];   // (64, 8194) f32
  const float* arg  = (const float*)d_in[1];   // (64, 8194) f32
  const float* cooc = (const float*)d_in[2];   // (8192, 8192, 3) f32
  float* out = (float*)d_out;                  // (64, 3) f32

  hipLaunchKernelGGL(cooc_mask_init, dim3(1), dim3(64), 0, stream, func, out);
  hipLaunchKernelGGL(cooc_einsum_wmma, dim3(V_DIM / 16), dim3(128), 0, stream,
                     func, arg, cooc, out);
}